// Non_localBlock_24077586661806
// MI455X (gfx1250) — compile-verified
//
#include <hip/hip_runtime.h>

typedef __attribute__((ext_vector_type(2))) float v2f;
typedef __attribute__((ext_vector_type(8))) float v8f;

#define WMMA_F32(a, b, c) \
  __builtin_amdgcn_wmma_f32_16x16x4_f32(false, (a), false, (b), (short)0, (c), false, false)

constexpr int kC  = 128;   // channels
constexpr int kHW = 4096;  // 64*64 pixels
constexpr int kP  = 1024;  // 32*32 pooled pixels

// ---------------------------------------------------------------------------
// k1: phi = maxpool2(w_phi @ x + b_phi), g = maxpool2(w_g @ x + b_g)
// Output layout: phi_ws/g_ws [B][64][1024] (channel-major, pooled row-major).
// Grid: (32 row-pairs, 16 batches), 256 threads (8 waves).
// ---------------------------------------------------------------------------
__global__ __launch_bounds__(256) void k1_conv_pool(
    const float* __restrict__ x, const float* __restrict__ w_phi,
    const float* __restrict__ b_phi, const float* __restrict__ w_g,
    const float* __restrict__ b_g, float* __restrict__ phi_ws,
    float* __restrict__ g_ws) {
  __shared__ float sWT[2][128][64];  // [mat][k][o]  (64 KB, transposed weights)

  const int b   = blockIdx.y;
  const int rp  = blockIdx.x;  // image row-pair (rows 2rp, 2rp+1)
  const int tid = threadIdx.x;

  // Load + transpose both weight matrices into LDS (conflict-free writes).
  for (int idx = tid; idx < 2 * 128 * 64; idx += 256) {
    const int m = idx >> 13;
    const int o = idx & 63;
    const int k = (idx >> 6) & 127;
    const float* w = m ? w_g : w_phi;
    sWT[m][k][o] = w[o * 128 + k];
  }
  __syncthreads();

  const int lane = tid & 31;
  const int wave = tid >> 5;
  const int ln = lane & 15;  // N (or M) position within tile
  const int lh = lane >> 4;  // half-wave -> K pair select
  const int mtile  = wave & 3;   // 4 out-channel tiles (64 ch)
  const int jgroup = wave >> 2;  // column half of the image row
  const int mo = mtile * 16 + ln;

  const float* xb = x + (size_t)b * kC * kHW;

  v8f z = {0.f, 0.f, 0.f, 0.f, 0.f, 0.f, 0.f, 0.f};
  v8f accP[2][2], accG[2][2];  // [col-subtile jj][image row]
#pragma unroll
  for (int jj = 0; jj < 2; ++jj)
#pragma unroll
    for (int r = 0; r < 2; ++r) { accP[jj][r] = z; accG[jj][r] = z; }

  const int h0 = rp * 2;
  for (int k = 0; k < 32; ++k) {  // K = 128, 4 per wmma
    const int k0 = 4 * k + 2 * lh;
    v2f aP, aG;
    aP.x = sWT[0][k0][mo];     aP.y = sWT[0][k0 + 1][mo];
    aG.x = sWT[1][k0][mo];     aG.y = sWT[1][k0 + 1][mo];
#pragma unroll
    for (int jj = 0; jj < 2; ++jj) {
      const int col = jgroup * 32 + jj * 16 + ln;
#pragma unroll
      for (int row = 0; row < 2; ++row) {
        const int pix = (h0 + row) * 64 + col;
        v2f bb;
        bb.x = xb[(size_t)k0 * kHW + pix];
        bb.y = xb[(size_t)(k0 + 1) * kHW + pix];
        accP[jj][row] = WMMA_F32(aP, bb, accP[jj][row]);
        accG[jj][row] = WMMA_F32(aG, bb, accG[jj][row]);
      }
    }
  }

  // 2x2 maxpool: vertical in registers, horizontal via lane^1 shuffle.
#pragma unroll
  for (int jj = 0; jj < 2; ++jj) {
#pragma unroll
    for (int r = 0; r < 8; ++r) {
      float vP = fmaxf(accP[jj][0][r], accP[jj][1][r]);
      float vG = fmaxf(accG[jj][0][r], accG[jj][1][r]);
      vP = fmaxf(vP, __shfl_xor(vP, 1, 32));
      vG = fmaxf(vG, __shfl_xor(vG, 1, 32));
      if ((ln & 1) == 0) {
        const int o  = mtile * 16 + r + 8 * lh;
        const int pw = (jgroup * 32 + jj * 16 + ln) >> 1;
        const int p  = rp * 32 + pw;
        phi_ws[((size_t)b * 64 + o) * kP + p] = vP + b_phi[o];
        g_ws  [((size_t)b * 64 + o) * kP + p] = vG + b_g[o];
      }
    }
  }
}

// ---------------------------------------------------------------------------
// k2: M[b] = phi[b] (64x1024) @ g[b]^T  -> 64x64 per batch (NT gemm, K=1024)
// Grid: 16 blocks (one per batch), 256 threads; each wave owns 2 output tiles.
// ---------------------------------------------------------------------------
__global__ __launch_bounds__(256) void k2_M(
    const float* __restrict__ phi_ws, const float* __restrict__ g_ws,
    float* __restrict__ M_ws) {
  __shared__ float sP[64][65];
  __shared__ float sG[64][65];
  const int b   = blockIdx.x;
  const int tid = threadIdx.x;
  const int lane = tid & 31, wave = tid >> 5;
  const int ln = lane & 15, lh = lane >> 4;
  const int t0 = wave * 2;
  const int ti = t0 >> 2, tj0 = t0 & 3;  // tiles (ti,tj0) and (ti,tj0+1)
  const float* pB = phi_ws + (size_t)b * 64 * kP;
  const float* gB = g_ws + (size_t)b * 64 * kP;

  v8f z = {0.f, 0.f, 0.f, 0.f, 0.f, 0.f, 0.f, 0.f};
  v8f acc0 = z, acc1 = z;

  for (int pb = 0; pb < 16; ++pb) {  // K blocks of 64
    for (int idx = tid; idx < 4096; idx += 256) {
      const int row = idx >> 6, col = idx & 63;
      sP[row][col] = pB[row * kP + pb * 64 + col];
      sG[row][col] = gB[row * kP + pb * 64 + col];
    }
    __syncthreads();
#pragma unroll
    for (int k = 0; k < 16; ++k) {
      const int p0 = 4 * k + 2 * lh;
      v2f a, b0, b1;
      a.x  = sP[ti * 16 + ln][p0];         a.y  = sP[ti * 16 + ln][p0 + 1];
      b0.x = sG[tj0 * 16 + ln][p0];        b0.y = sG[tj0 * 16 + ln][p0 + 1];
      b1.x = sG[(tj0 + 1) * 16 + ln][p0];  b1.y = sG[(tj0 + 1) * 16 + ln][p0 + 1];
      acc0 = WMMA_F32(a, b0, acc0);
      acc1 = WMMA_F32(a, b1, acc1);
    }
    __syncthreads();
  }
#pragma unroll
  for (int r = 0; r < 8; ++r) {
    const int i = ti * 16 + r + 8 * lh;
    M_ws[(size_t)b * 4096 + i * 64 + tj0 * 16 + ln]       = acc0[r];
    M_ws[(size_t)b * 4096 + i * 64 + (tj0 + 1) * 16 + ln] = acc1[r];
  }
}

// ---------------------------------------------------------------------------
// k3: fold into per-batch affine map:
//   AT[b][ci][o] = (w_out @ M[b]^T @ w_theta)[o][ci]   (stored k-major!)
//   v[b][o]      = (w_out @ M[b]^T @ b_theta)[o] + b_out[o]
// Tiny (50 MFLOP total) -> plain VALU, one block per batch.
// ---------------------------------------------------------------------------
__global__ __launch_bounds__(256) void k3_fold(
    const float* __restrict__ M_ws, const float* __restrict__ w_theta,
    const float* __restrict__ b_theta, const float* __restrict__ w_out,
    const float* __restrict__ b_out, float* __restrict__ AT_ws,
    float* __restrict__ v_ws) {
  __shared__ float sM[64][65];   // M[i][j]
  __shared__ float sT[64][128];  // T[j][ci] = sum_i M[i][j] * w_theta[i][ci]
  __shared__ float sTb[64];
  const int b   = blockIdx.x;
  const int tid = threadIdx.x;
  for (int idx = tid; idx < 4096; idx += 256)
    sM[idx >> 6][idx & 63] = M_ws[(size_t)b * 4096 + idx];
  __syncthreads();
  for (int idx = tid; idx < 8192; idx += 256) {
    const int j = idx >> 7, ci = idx & 127;
    float s = 0.f;
    for (int i = 0; i < 64; ++i) s += sM[i][j] * w_theta[i * 128 + ci];
    sT[j][ci] = s;
  }
  if (tid < 64) {
    float s = 0.f;
    for (int i = 0; i < 64; ++i) s += sM[i][tid] * b_theta[i];
    sTb[tid] = s;
  }
  __syncthreads();
  for (int idx = tid; idx < 16384; idx += 256) {
    const int o = idx & 127, ci = idx >> 7;
    float s = 0.f;
    for (int j = 0; j < 64; ++j) s += w_out[o * 64 + j] * sT[j][ci];
    AT_ws[(size_t)b * 16384 + ci * 128 + o] = s;  // transposed store
  }
  if (tid < 128) {
    float s = b_out[tid];
    for (int j = 0; j < 64; ++j) s += w_out[tid * 64 + j] * sTb[j];
    v_ws[(size_t)b * 128 + tid] = s;
  }
}

// ---------------------------------------------------------------------------
// k4: y[b] = x[b] + A[b] @ x[b] + v[b]   (GEMM M=128,K=128,N=4096 per batch)
// Grid: (32 pixel-blocks of 128, 16 batches); X block staged once in LDS;
// each wave keeps 8 accumulator tiles so A-fragments load once per k-step.
// ---------------------------------------------------------------------------
__global__ __launch_bounds__(256) void k4_apply(
    const float* __restrict__ x, const float* __restrict__ AT_ws,
    const float* __restrict__ v_ws, float* __restrict__ out) {
  __shared__ float sX[128][128];  // 64 KB
  const int b = blockIdx.y, pb = blockIdx.x;
  const int tid = threadIdx.x;
  const float* xb = x + (size_t)b * kC * kHW;
  const int pxbase = pb * 128;
  for (int idx = tid; idx < 16384; idx += 256) {
    const int ch = idx >> 7, col = idx & 127;
    sX[ch][col] = xb[(size_t)ch * kHW + pxbase + col];
  }
  __syncthreads();

  const int lane = tid & 31, wave = tid >> 5;
  const int ln = lane & 15, lh = lane >> 4;
  const int ob = wave * 16;  // 16 out-channels per wave
  const float* ATb = AT_ws + (size_t)b * 16384;

  v8f z = {0.f, 0.f, 0.f, 0.f, 0.f, 0.f, 0.f, 0.f};
  v8f acc[8];
#pragma unroll
  for (int nt = 0; nt < 8; ++nt) acc[nt] = z;

  for (int k = 0; k < 32; ++k) {
    const int k0 = 4 * k + 2 * lh;
    v2f a;
    a.x = ATb[k0 * 128 + ob + ln];        // coalesced: AT is k-major
    a.y = ATb[(k0 + 1) * 128 + ob + ln];
#pragma unroll
    for (int nt = 0; nt < 8; ++nt) {
      v2f bb;
      bb.x = sX[k0][nt * 16 + ln];
      bb.y = sX[k0 + 1][nt * 16 + ln];
      acc[nt] = WMMA_F32(a, bb, acc[nt]);
    }
  }

  const float* vb = v_ws + (size_t)b * 128;
  float* outb = out + (size_t)b * kC * kHW;
#pragma unroll
  for (int nt = 0; nt < 8; ++nt) {
#pragma unroll
    for (int r = 0; r < 8; ++r) {
      const int o = ob + r + 8 * lh;
      const int col = nt * 16 + ln;
      outb[(size_t)o * kHW + pxbase + col] = acc[nt][r] + sX[o][col] + vb[o];
    }
  }
}

// ---------------------------------------------------------------------------
extern "C" void kernel_launch(void* const* d_in, const int* in_sizes, int n_in,
                              void* d_out, int out_size, void* d_ws,
                              size_t ws_size, hipStream_t stream) {
  const float* x       = (const float*)d_in[0];
  const float* w_theta = (const float*)d_in[1];
  const float* b_theta = (const float*)d_in[2];
  const float* w_phi   = (const float*)d_in[3];
  const float* b_phi   = (const float*)d_in[4];
  const float* w_g     = (const float*)d_in[5];
  const float* b_g     = (const float*)d_in[6];
  const float* w_out   = (const float*)d_in[7];
  const float* b_out   = (const float*)d_in[8];
  float* out = (float*)d_out;

  char* ws = (char*)d_ws;
  float* phi_ws = (float*)(ws);                 // 16*64*1024*4 = 4 MB
  float* g_ws   = (float*)(ws + 4194304);       // 4 MB
  float* M_ws   = (float*)(ws + 8388608);       // 256 KB
  float* AT_ws  = (float*)(ws + 8650752);       // 1 MB
  float* v_ws   = (float*)(ws + 9699328);       // 8 KB

  k1_conv_pool<<<dim3(32, 16), 256, 0, stream>>>(x, w_phi, b_phi, w_g, b_g,
                                                 phi_ws, g_ws);
  k2_M<<<dim3(16), 256, 0, stream>>>(phi_ws, g_ws, M_ws);
  k3_fold<<<dim3(16), 256, 0, stream>>>(M_ws, w_theta, b_theta, w_out, b_out,
                                        AT_ws, v_ws);
  k4_apply<<<dim3(32, 16), 256, 0, stream>>>(x, AT_ws, v_ws, out);
}